// Net_2284922601977
// MI455X (gfx1250) — compile-verified
//
#include <hip/hip_runtime.h>
#include <hip/hip_bf16.h>

// ---------------- problem constants ----------------
#define BG   8
#define NN   4096
#define BN   (BG*NN)        // 32768
#define EPB  65536          // edges per graph
#define ET   (BG*EPB)       // 524288 total edges
#define FIN  128
#define HID  512
#define CC   2
#define NOUT 10

// output layout (floats): [logits 80][mc 1][o 1][s_logits 65536][out_adj 32]
#define OFF_LOGITS 0
#define OFF_MC     80
#define OFF_O      81
#define OFF_SLOG   82
#define OFF_ADJ    (82 + BN*CC)   // 65618

// workspace layout (floats)
#define WS_AGG   0                              // BN*FIN           = 4194304
#define WS_DEG   (WS_AGG + BN*FIN)              // BN               = 32768
#define WS_H     (WS_DEG + BN)                  // BN*HID           = 16777216
#define WS_S     (WS_H + BN*HID)                // BN*CC
#define WS_POOL  (WS_S + BN*CC)                 // BG*CC*HID        = 8192
#define WS_RAW   (WS_POOL + BG*CC*HID)          // BG*4
#define WS_SS    (WS_RAW + BG*4)                // BG*3
#define WS_DEN   (WS_SS + BG*3)                 // BG
#define WS_ADJN  (WS_DEN + BG)                  // BG*4
#define WS_XM    (WS_ADJN + BG*4)               // BG*HID
#define WS_H2    (WS_XM + BG*HID)               // BG*HID

typedef __attribute__((ext_vector_type(2))) float v2f;
typedef __attribute__((ext_vector_type(8))) float v8f;

// ---------------- 1) edge gather/scatter: agg[dst] += x[src]*w ; deg[src] += w
__global__ __launch_bounds__(256)
void edge_agg_kernel(const int* __restrict__ ei, const float* __restrict__ ew,
                     const float* __restrict__ x, float* __restrict__ agg,
                     float* __restrict__ deg)
{
    int gwave = (blockIdx.x * blockDim.x + threadIdx.x) >> 5;
    int lane  = threadIdx.x & 31;
    if (gwave >= ET) return;
    int   src = ei[gwave];
    int   dst = ei[ET + gwave];
    float w   = ew[gwave];
    const float* xs = x   + src * FIN;
    float*       ad = agg + dst * FIN;
#pragma unroll
    for (int i = 0; i < 4; ++i) {
        int f = lane + 32 * i;
        atomicAdd(&ad[f], xs[f] * w);
    }
    if (lane == 0) atomicAdd(&deg[src], w);
}

// ---------------- 2) h = relu(agg@Wrel1 + x@Wroot1 + brel1) via f32 WMMA 16x16x4
//  A tiles (16x128 of agg and x) are staged in LDS with coalesced global loads;
//  row pitch padded to 130 floats -> conflict-free ds_load_b64 A-fragments.
#define APITCH 130
__global__ __launch_bounds__(256)
void gemm_h_kernel(const float* __restrict__ agg, const float* __restrict__ x,
                   const float* __restrict__ Wrel, const float* __restrict__ Wroot,
                   const float* __restrict__ brel, float* __restrict__ h)
{
    __shared__ float shA[16 * APITCH];   // agg tile
    __shared__ float shX[16 * APITCH];   // x tile

    const int lane = threadIdx.x & 31;
    const int wave = threadIdx.x >> 5;
    const int m0   = blockIdx.x * 16;                 // 2048 m-tiles
    const int n0   = blockIdx.y * 128 + wave * 16;    // 4 * 8 waves = 32 n-tiles
    const int half = lane >> 4;                       // 0 -> K{0,1}, 1 -> K{2,3}
    const int l15  = lane & 15;
    const int bcol = n0 + l15;

    // cooperative, coalesced tile load: 16 rows x 128 K each, 8 elems/thread
    for (int t = threadIdx.x; t < 16 * FIN; t += 256) {
        int r = t >> 7;          // row in tile
        int k = t & (FIN - 1);   // K index
        shA[r * APITCH + k] = agg[(m0 + r) * FIN + k];
        shX[r * APITCH + k] = x[(m0 + r) * FIN + k];
    }
    __syncthreads();

    const float* arow = shA + l15 * APITCH;
    const float* xrow = shX + l15 * APITCH;

    v8f acc = {};
    // pass 1: agg @ Wrel1  (K = 128)
    for (int k = 0; k < FIN; k += 4) {
        int ka = k + half * 2;
        v2f a, b;
        a.x = arow[ka];
        a.y = arow[ka + 1];
        b.x = Wrel[ka * HID + bcol];
        b.y = Wrel[(ka + 1) * HID + bcol];
        acc = __builtin_amdgcn_wmma_f32_16x16x4_f32(false, a, false, b,
                                                    (short)0, acc, false, false);
    }
    // pass 2: x @ Wroot1  (K = 128), same accumulator
    for (int k = 0; k < FIN; k += 4) {
        int ka = k + half * 2;
        v2f a, b;
        a.x = xrow[ka];
        a.y = xrow[ka + 1];
        b.x = Wroot[ka * HID + bcol];
        b.y = Wroot[(ka + 1) * HID + bcol];
        acc = __builtin_amdgcn_wmma_f32_16x16x4_f32(false, a, false, b,
                                                    (short)0, acc, false, false);
    }
    // C/D layout: VGPR r -> row m0 + r + 8*half ; col = n0 + (lane&15)
    float bias = brel[bcol];
    int rowbase = m0 + half * 8;
#pragma unroll
    for (int r = 0; r < 8; ++r) {
        float v = acc[r] + bias;
        v = v > 0.f ? v : 0.f;
        h[(rowbase + r) * HID + bcol] = v;
    }
}

// ---------------- 3) per-node: s_logits, softmax s, pooled sums, den, ss
__global__ __launch_bounds__(256)
void node_kernel(const float* __restrict__ h, const float* __restrict__ Wp,
                 const float* __restrict__ bp, const float* __restrict__ deg,
                 float* __restrict__ s, float* __restrict__ slog_out,
                 float* __restrict__ pool, float* __restrict__ ss,
                 float* __restrict__ den)
{
    __shared__ float lds_pool[CC * HID];
    __shared__ float lds_ss[3];
    __shared__ float lds_den;
    for (int t = threadIdx.x; t < CC * HID; t += 256) lds_pool[t] = 0.f;
    if (threadIdx.x < 3) lds_ss[threadIdx.x] = 0.f;
    if (threadIdx.x == 0) lds_den = 0.f;
    __syncthreads();

    const int lane = threadIdx.x & 31;
    const int wave = threadIdx.x >> 5;
    float wp0[16], wp1[16];
#pragma unroll
    for (int i = 0; i < 16; ++i) {
        int f = lane + 32 * i;
        wp0[i] = Wp[f * CC + 0];
        wp1[i] = Wp[f * CC + 1];
    }
    float bp0 = bp[0], bp1 = bp[1];
    int n0 = blockIdx.x * 512 + wave * 64;

    for (int j = 0; j < 64; ++j) {
        int n = n0 + j;
        float hv[16];
        float d0 = 0.f, d1 = 0.f;
#pragma unroll
        for (int i = 0; i < 16; ++i) {
            hv[i] = h[n * HID + lane + 32 * i];
            d0 += hv[i] * wp0[i];
            d1 += hv[i] * wp1[i];
        }
        // butterfly reduce -> every lane holds the totals
        for (int off = 16; off > 0; off >>= 1) {
            d0 += __shfl_xor(d0, off, 32);
            d1 += __shfl_xor(d1, off, 32);
        }
        float l0 = d0 + bp0, l1 = d1 + bp1;
        float m  = fmaxf(l0, l1);
        float e0 = expf(l0 - m), e1 = expf(l1 - m);
        float inv = 1.f / (e0 + e1);
        float s0 = e0 * inv, s1 = e1 * inv;
        if (lane == 0) {
            slog_out[n * CC + 0] = l0;
            slog_out[n * CC + 1] = l1;
            s[n * CC + 0] = s0;
            s[n * CC + 1] = s1;
            float dn = deg[n];
            atomicAdd(&lds_den, dn * (s0 * s0 + s1 * s1));
            atomicAdd(&lds_ss[0], s0 * s0);
            atomicAdd(&lds_ss[1], s0 * s1);
            atomicAdd(&lds_ss[2], s1 * s1);
        }
#pragma unroll
        for (int i = 0; i < 16; ++i) {
            int f = lane + 32 * i;
            atomicAdd(&lds_pool[f],       s0 * hv[i]);
            atomicAdd(&lds_pool[HID + f], s1 * hv[i]);
        }
    }
    __syncthreads();
    int b = blockIdx.x >> 3;   // 8 blocks of 512 nodes per graph
    for (int t = threadIdx.x; t < CC * HID; t += 256)
        atomicAdd(&pool[b * CC * HID + t], lds_pool[t]);
    if (threadIdx.x < 3) atomicAdd(&ss[b * 3 + threadIdx.x], lds_ss[threadIdx.x]);
    if (threadIdx.x == 0) atomicAdd(&den[b], lds_den);
}

// ---------------- 4) per-edge: raw out_adj[b] = s^T A s  (2x2 per batch)
__global__ __launch_bounds__(256)
void edge_adj_kernel(const int* __restrict__ ei, const float* __restrict__ ew,
                     const float* __restrict__ s, float* __restrict__ raw)
{
    int b = blockIdx.x >> 4;   // 16 blocks of 4096 edges per graph
    float a00 = 0.f, a01 = 0.f, a10 = 0.f, a11 = 0.f;
    int base = blockIdx.x * 4096;
    for (int t = threadIdx.x; t < 4096; t += 256) {
        int e = base + t;
        int src = ei[e];
        int dst = ei[ET + e];
        float w = ew[e];
        float r0 = s[src * CC], r1 = s[src * CC + 1];
        float c0 = s[dst * CC], c1 = s[dst * CC + 1];
        a00 += w * r0 * c0; a01 += w * r0 * c1;
        a10 += w * r1 * c0; a11 += w * r1 * c1;
    }
    __shared__ float red[4];
    if (threadIdx.x < 4) red[threadIdx.x] = 0.f;
    __syncthreads();
    atomicAdd(&red[0], a00); atomicAdd(&red[1], a01);
    atomicAdd(&red[2], a10); atomicAdd(&red[3], a11);
    __syncthreads();
    if (threadIdx.x < 4) atomicAdd(&raw[b * 4 + threadIdx.x], red[threadIdx.x]);
}

// ---------------- 5) losses + normalized out_adj
__global__ __launch_bounds__(64)
void finalize_kernel(const float* __restrict__ raw, const float* __restrict__ ss,
                     const float* __restrict__ den, float* __restrict__ adjn,
                     float* __restrict__ out)
{
    __shared__ float mcs[BG], osv[BG];
    int b = threadIdx.x;
    if (b < BG) {
        float num = raw[4 * b + 0] + raw[4 * b + 3];
        mcs[b] = -(num / den[b]);
        float s00 = ss[3 * b], s01 = ss[3 * b + 1], s11 = ss[3 * b + 2];
        float nrm = sqrtf(s00 * s00 + 2.f * s01 * s01 + s11 * s11);
        const float iv = 0.70710678118654752f; // 1/sqrt(2)
        float d00 = s00 / nrm - iv, d01 = s01 / nrm, d11 = s11 / nrm - iv;
        osv[b] = sqrtf(d00 * d00 + 2.f * d01 * d01 + d11 * d11);
        // zero diagonal + symmetric degree normalization
        float a01 = raw[4 * b + 1], a10 = raw[4 * b + 2];
        float d2_0 = sqrtf(a01) + 1e-15f;
        float d2_1 = sqrtf(a10) + 1e-15f;
        float n01 = a01 / (d2_0 * d2_1);
        float n10 = a10 / (d2_1 * d2_0);
        adjn[4 * b + 0] = 0.f; adjn[4 * b + 1] = n01;
        adjn[4 * b + 2] = n10; adjn[4 * b + 3] = 0.f;
        out[OFF_ADJ + 4 * b + 0] = 0.f; out[OFF_ADJ + 4 * b + 1] = n01;
        out[OFF_ADJ + 4 * b + 2] = n10; out[OFF_ADJ + 4 * b + 3] = 0.f;
    }
    __syncthreads();
    if (threadIdx.x == 0) {
        float mc = 0.f, ol = 0.f;
        for (int i = 0; i < BG; ++i) { mc += mcs[i]; ol += osv[i]; }
        out[OFF_MC] = mc / BG;
        out[OFF_O]  = ol / BG;
    }
}

// ---------------- 6) pooled conv3 + mean over C -> xm[b, HID]
__global__ __launch_bounds__(256)
void xm_kernel(const float* __restrict__ adjn, const float* __restrict__ pool,
               const float* __restrict__ Wrel3, const float* __restrict__ brel3,
               const float* __restrict__ Wroot3, float* __restrict__ xm)
{
    int idx = blockIdx.x * 256 + threadIdx.x;   // 16 blocks -> 4096 threads
    int b = idx >> 9;
    int o = idx & (HID - 1);
    float a00 = adjn[4 * b + 0], a01 = adjn[4 * b + 1];
    float a10 = adjn[4 * b + 2], a11 = adjn[4 * b + 3];
    const float* p0 = pool + b * CC * HID;
    const float* p1 = p0 + HID;
    float acc0 = 0.f, acc1 = 0.f;
    for (int k = 0; k < HID; ++k) {
        float wr = Wrel3[k * HID + o];
        float wt = Wroot3[k * HID + o];
        float o0 = p0[k], o1 = p1[k];
        float g0 = a00 * o0 + a10 * o1;   // agg3[b,0,k] = sum_i A[i,0]*out[i,k]
        float g1 = a01 * o0 + a11 * o1;   // agg3[b,1,k]
        acc0 += g0 * wr + o0 * wt;
        acc1 += g1 * wr + o1 * wt;
    }
    xm[b * HID + o] = 0.5f * (acc0 + acc1) + brel3[o];
}

// ---------------- 7) h2 = relu(xm @ Wlin1 + blin1)
__global__ __launch_bounds__(256)
void lin1_kernel(const float* __restrict__ xm, const float* __restrict__ Wlin1,
                 const float* __restrict__ blin1, float* __restrict__ h2)
{
    int idx = blockIdx.x * 256 + threadIdx.x;
    int b = idx >> 9;
    int o = idx & (HID - 1);
    float acc = blin1[o];
    const float* xr = xm + b * HID;
    for (int k = 0; k < HID; ++k) acc += xr[k] * Wlin1[k * HID + o];
    h2[b * HID + o] = acc > 0.f ? acc : 0.f;
}

// ---------------- 8) logits + log_softmax
__global__ __launch_bounds__(128)
void logits_kernel(const float* __restrict__ h2, const float* __restrict__ Wlin2,
                   const float* __restrict__ blin2, float* __restrict__ out)
{
    __shared__ float lg[BG * NOUT];
    int t = threadIdx.x;
    if (t < BG * NOUT) {
        int b = t / NOUT, o = t % NOUT;
        float acc = blin2[o];
        const float* hr = h2 + b * HID;
        for (int k = 0; k < HID; ++k) acc += hr[k] * Wlin2[k * NOUT + o];
        lg[t] = acc;
    }
    __syncthreads();
    if (t < BG) {
        float m = -1e30f;
        for (int o = 0; o < NOUT; ++o) m = fmaxf(m, lg[t * NOUT + o]);
        float sum = 0.f;
        for (int o = 0; o < NOUT; ++o) sum += expf(lg[t * NOUT + o] - m);
        float lse = m + logf(sum);
        for (int o = 0; o < NOUT; ++o)
            out[OFF_LOGITS + t * NOUT + o] = lg[t * NOUT + o] - lse;
    }
}

// ---------------- launcher ----------------
extern "C" void kernel_launch(void* const* d_in, const int* in_sizes, int n_in,
                              void* d_out, int out_size, void* d_ws, size_t ws_size,
                              hipStream_t stream)
{
    const float* x      = (const float*)d_in[0];
    const int*   ei     = (const int*)  d_in[1];
    const float* ew     = (const float*)d_in[3];
    const float* Wrel1  = (const float*)d_in[4];
    const float* brel1  = (const float*)d_in[5];
    const float* Wroot1 = (const float*)d_in[6];
    const float* Wpool  = (const float*)d_in[7];
    const float* bpool  = (const float*)d_in[8];
    const float* Wrel3  = (const float*)d_in[9];
    const float* brel3  = (const float*)d_in[10];
    const float* Wroot3 = (const float*)d_in[11];
    const float* Wlin1  = (const float*)d_in[12];
    const float* blin1  = (const float*)d_in[13];
    const float* Wlin2  = (const float*)d_in[14];
    const float* blin2  = (const float*)d_in[15];
    float* out = (float*)d_out;
    float* ws  = (float*)d_ws;

    float* agg  = ws + WS_AGG;
    float* deg  = ws + WS_DEG;
    float* h    = ws + WS_H;
    float* s    = ws + WS_S;
    float* pool = ws + WS_POOL;
    float* raw  = ws + WS_RAW;
    float* ssb  = ws + WS_SS;
    float* den  = ws + WS_DEN;
    float* adjn = ws + WS_ADJN;
    float* xm   = ws + WS_XM;
    float* h2   = ws + WS_H2;

    // zero accumulators (graph-capture-safe)
    hipMemsetAsync(agg,  0, (size_t)(BN * FIN + BN) * sizeof(float), stream);
    hipMemsetAsync(pool, 0, (size_t)(BG * CC * HID + BG * 4 + BG * 3 + BG) * sizeof(float), stream);

    edge_agg_kernel<<<ET / 8, 256, 0, stream>>>(ei, ew, x, agg, deg);
    gemm_h_kernel<<<dim3(BN / 16, HID / 128), 256, 0, stream>>>(agg, x, Wrel1, Wroot1, brel1, h);
    node_kernel<<<BN / 512, 256, 0, stream>>>(h, Wpool, bpool, deg, s, out + OFF_SLOG,
                                              pool, ssb, den);
    edge_adj_kernel<<<ET / 4096, 256, 0, stream>>>(ei, ew, s, raw);
    finalize_kernel<<<1, 64, 0, stream>>>(raw, ssb, den, adjn, out);
    xm_kernel<<<BG * HID / 256, 256, 0, stream>>>(adjn, pool, Wrel3, brel3, Wroot3, xm);
    lin1_kernel<<<BG * HID / 256, 256, 0, stream>>>(xm, Wlin1, blin1, h2);
    logits_kernel<<<1, 128, 0, stream>>>(h2, Wlin2, blin2, out);
}